// ActSWL_80058190397913
// MI455X (gfx1250) — compile-verified
//
#include <hip/hip_runtime.h>

// ActSWL: out_t = silu(cumsum_t(x)) - silu(cumsum_{t-1}(x)), T=4 recurrence in
// registers. Pure HBM-streaming problem: 512 MB round trip @ 23.3 TB/s ~ 22 us.
// CDNA5 path: clause of 4x global_load_b128 th:TH_LOAD_NT (input > 192 MB L2,
// touched once), packed-f32 VALU, v_exp_f32 + v_rcp_f32 SiLU (no precise-div
// lowering), NT b128 stores, wave32-dense grid.

typedef float v4f __attribute__((ext_vector_type(4)));

#ifndef ACTSWL_T
#define ACTSWL_T 4
#endif

__device__ __forceinline__ float silu_f32(float a) {
    // a * sigmoid(a) = a * rcp(1 + exp(-a)).
    // v_exp_f32 (with -log2e pre-mul folded by compiler) + v_rcp_f32: 1-ulp,
    // 5 VALU ops total vs ~12 for the IEEE divide lowering.
    // a -> large negative: exp(-a)=inf, rcp(inf)=0, result -0 (matches divide
    // path). NaN propagates.
    float e = __expf(-a);
    float s = __builtin_amdgcn_rcpf(1.0f + e);
    return a * s;
}

__global__ __launch_bounds__(256) void actswl_kernel(
    const float* __restrict__ x, float* __restrict__ out, unsigned inner4)
{
    unsigned i = blockIdx.x * blockDim.x + threadIdx.x;  // float4 index into one t-plane
    if (i >= inner4) return;

    const v4f* __restrict__ xin = (const v4f*)x;
    v4f* __restrict__ o = (v4f*)out;

    // All four t-plane loads are independent (t-stride = 64 MB, guaranteed
    // L2 miss): issue them as one clause, non-temporal so they stream past L2.
    v4f v0 = __builtin_nontemporal_load(&xin[i + 0u * inner4]);
    v4f v1 = __builtin_nontemporal_load(&xin[i + 1u * inner4]);
    v4f v2 = __builtin_nontemporal_load(&xin[i + 2u * inner4]);
    v4f v3 = __builtin_nontemporal_load(&xin[i + 3u * inner4]);
    v4f vt[ACTSWL_T] = {v0, v1, v2, v3};

    v4f acc;   acc[0] = acc[1] = acc[2] = acc[3] = 0.0f;
    v4f yprev; yprev[0] = yprev[1] = yprev[2] = yprev[3] = 0.0f;

#pragma unroll
    for (int t = 0; t < ACTSWL_T; ++t) {
        acc += vt[t];
        v4f y;
#pragma unroll
        for (int c = 0; c < 4; ++c) y[c] = silu_f32(acc[c]);
        v4f d = y - yprev;
        __builtin_nontemporal_store(d, &o[i + (unsigned)t * inner4]);
        yprev = y;
    }
}

extern "C" void kernel_launch(void* const* d_in, const int* in_sizes, int n_in,
                              void* d_out, int out_size, void* d_ws, size_t ws_size,
                              hipStream_t stream) {
    (void)n_in; (void)out_size; (void)d_ws; (void)ws_size;
    const float* x = (const float*)d_in[0];
    float* out = (float*)d_out;

    // in_sizes[0] = T*B*L*D; inner plane = B*L*D; D=4096 so /4 is exact.
    const long long total = (long long)in_sizes[0];
    const unsigned inner  = (unsigned)(total / ACTSWL_T);
    const unsigned inner4 = inner / 4u;

    const unsigned block = 256u;
    const unsigned grid  = (inner4 + block - 1u) / block;
    actswl_kernel<<<grid, block, 0, stream>>>(x, out, inner4);
}